// CuEquivarianceLayer_21500606283806
// MI455X (gfx1250) — compile-verified
//
#include <hip/hip_runtime.h>
#include <cmath>
#include <complex>
#include <algorithm>

typedef float v2f __attribute__((ext_vector_type(2)));
typedef float v8f __attribute__((ext_vector_type(8)));

// CG/alpha coefficients passed by value to kernels (POD, graph-capture safe).
struct Coefs {
    float c0, c1, c2, c3;      // alpha-folded scalars for paths 0..3 (identity CGs)
    float cg4[3][5][3];        // alpha-folded real CG for path 4: 1 x 2 -> 1
};

// ---------------------------------------------------------------------------
// Host: real Clebsch-Gordan tensors (line-for-line port of the reference).
// ---------------------------------------------------------------------------
namespace {

using cd = std::complex<double>;

double factd(int n) { double r = 1.0; for (int i = 2; i <= n; ++i) r *= i; return r; }

// SU(2) CG via Racah formula; C has dims (2j1+1)x(2j2+1)x(2j3+1), row-major.
void su2_cg(int j1, int j2, int j3, double* C) {
    const int d2 = 2 * j2 + 1, d3 = 2 * j3 + 1, d1 = 2 * j1 + 1;
    for (int i = 0; i < d1 * d2 * d3; ++i) C[i] = 0.0;
    for (int m1 = -j1; m1 <= j1; ++m1) {
        for (int m2 = -j2; m2 <= j2; ++m2) {
            int m3 = m1 + m2;
            if (m3 < -j3 || m3 > j3) continue;
            int vmin = std::max(std::max(-j1 + j2 + m3, -j1 + m1), 0);
            int vmax = std::min(std::min(j2 + j3 + m1, j3 - j1 + j2), j3 + m3);
            double c = std::sqrt((2.0 * j3 + 1.0) * factd(j3 + j1 - j2) * factd(j3 - j1 + j2)
                                 * factd(j1 + j2 - j3) / factd(j1 + j2 + j3 + 1));
            c *= std::sqrt(factd(j3 + m3) * factd(j3 - m3) * factd(j1 - m1)
                           * factd(j1 + m1) * factd(j2 - m2) * factd(j2 + m2));
            double s = 0.0;
            for (int v = vmin; v <= vmax; ++v) {
                double sign = ((v + j2 + m2) & 1) ? -1.0 : 1.0;
                s += sign * factd(j2 + j3 + m1 - v) * factd(j1 - m1 + v)
                     / (factd(v) * factd(j3 - j1 + j2 - v) * factd(j3 + m3 - v)
                        * factd(v + j1 - j2 - m3));
            }
            C[((j1 + m1) * d2 + (j2 + m2)) * d3 + (j3 + m3)] = c * s;
        }
    }
}

// Real->complex change-of-basis matrix (e3nn convention), (2l+1)x(2l+1).
void qmat(int l, cd* q) {
    const int d = 2 * l + 1;
    for (int i = 0; i < d * d; ++i) q[i] = cd(0, 0);
    const double is2 = 1.0 / std::sqrt(2.0);
    for (int m = -l; m < 0; ++m) {
        q[(l + m) * d + (l - m)] = cd(is2, 0.0);    // col l+|m|
        q[(l + m) * d + (l + m)] = cd(0.0, -is2);   // col l-|m|
    }
    q[l * d + l] = cd(1.0, 0.0);
    for (int m = 1; m <= l; ++m) {
        double sgn = (m & 1) ? -1.0 : 1.0;
        q[(l + m) * d + (l + m)] = cd(sgn * is2, 0.0);
        q[(l + m) * d + (l - m)] = cd(0.0, sgn * is2);
    }
    cd f(1.0, 0.0), mi(0.0, -1.0);
    for (int t = 0; t < l; ++t) f *= mi;            // (-i)^l
    for (int i = 0; i < d * d; ++i) q[i] *= f;
}

// Real CG tensor, unit Frobenius norm. out is d1*d2*d3 floats (l<=2).
void real_cg(int l1, int l2, int l3, float* out) {
    const int d1 = 2 * l1 + 1, d2 = 2 * l2 + 1, d3 = 2 * l3 + 1;
    double C[125];
    su2_cg(l1, l2, l3, C);
    cd q1[25], q2[25], q3[25];
    qmat(l1, q1); qmat(l2, q2); qmat(l3, q3);
    double R[75];
    double norm2 = 0.0;
    for (int j = 0; j < d1; ++j)
        for (int p = 0; p < d2; ++p)
            for (int m = 0; m < d3; ++m) {
                cd acc(0.0, 0.0);
                for (int i = 0; i < d1; ++i)
                    for (int k = 0; k < d2; ++k)
                        for (int n = 0; n < d3; ++n) {
                            double cv = C[(i * d2 + k) * d3 + n];
                            if (cv == 0.0) continue;
                            acc += q1[i * d1 + j] * q2[k * d2 + p] * std::conj(q3[n * d3 + m]) * cv;
                        }
                R[(j * d2 + p) * d3 + m] = acc.real();
                norm2 += acc.real() * acc.real();
            }
    double inv = 1.0 / std::sqrt(norm2);
    for (int i = 0; i < d1 * d2 * d3; ++i) out[i] = (float)(R[i] * inv);
}

Coefs make_coefs() {
    float cg000[1], cg011[9], cg101[9], cg110[9], cg121[45];
    real_cg(0, 0, 0, cg000);
    real_cg(0, 1, 1, cg011);
    real_cg(1, 0, 1, cg101);
    real_cg(1, 1, 0, cg110);
    real_cg(1, 2, 1, cg121);
    // alpha = 1/sqrt(u*v*N_paths_to_out); out0 has 2 paths, out1 has 3.
    const double a2 = 1.0 / std::sqrt(32.0 * 1.0 * 2.0);
    const double a3 = 1.0 / std::sqrt(32.0 * 1.0 * 3.0);
    Coefs cf;
    cf.c0 = (float)(a2 * (double)cg000[0]);   // path0: 0x0->0
    cf.c1 = (float)(a3 * (double)cg011[0]);   // path1: 0x1->1 (identity * c)
    cf.c2 = (float)(a3 * (double)cg101[0]);   // path2: 1x0->1 (identity * c)
    cf.c3 = (float)(a2 * (double)cg110[0]);   // path3: 1x1->0 (dot * c)
    for (int i = 0; i < 3; ++i)
        for (int j = 0; j < 5; ++j)
            for (int k = 0; k < 3; ++k)
                cf.cg4[i][j][k] = (float)(a3 * (double)cg121[(i * 5 + j) * 3 + k]);
    return cf;
}

} // namespace

// ---------------------------------------------------------------------------
// Kernel 0: zero the output (harness poisons d_out).
// ---------------------------------------------------------------------------
__global__ void eq_zero_out(float* __restrict__ out, int n) {
    int i = blockIdx.x * blockDim.x + threadIdx.x;
    if (i < n) out[i] = 0.0f;
}

// ---------------------------------------------------------------------------
// Kernel 1: node-level weight application via f32 WMMA.
// z layout: [node][chunk][w], chunk in [0,11), w in [0,32):
//   chunk 0   : sum_u x0[n,u] * W0[u,w]
//   chunk 1   : sum_u x0[n,u] * W1[u,w]
//   chunk 2+i : sum_u x1[n,u,i] * W2[u,w]   (i = 0..2)
//   chunk 5+i : sum_u x1[n,u,i] * W3[u,w]
//   chunk 8+i : sum_u x1[n,u,i] * W4[u,w]
// One wave per (chunk, 16-w tile, group of TPW 16-node tiles); B (weights)
// loaded once per wave and reused across tiles; 8x V_WMMA_F32_16X16X4_F32
// per tile.
// ---------------------------------------------------------------------------
#define EQ_TPW 4   // node tiles per wave

__global__ void eq_precompute_z(const float* __restrict__ features,
                                const float* __restrict__ weights,
                                float* __restrict__ z, int num_nodes) {
    const int tgrp  = blockIdx.x;          // group of EQ_TPW 16-node tiles
    const int task  = blockIdx.y;          // chunk*2 + wtile, 22 tasks
    const int chunk = task >> 1;
    const int wtile = task & 1;
    const int lane  = threadIdx.x;         // 0..31
    const int lo    = lane & 15;
    const int hi    = lane >> 4;           // 0 or 1
    const int ntiles = (num_nodes + 15) >> 4;

    int p, base, stride;
    if (chunk < 2)      { p = chunk; base = 0;                stride = 1; }
    else if (chunk < 5) { p = 2;     base = 32 + (chunk - 2); stride = 3; }
    else if (chunk < 8) { p = 3;     base = 32 + (chunk - 5); stride = 3; }
    else                { p = 4;     base = 32 + (chunk - 8); stride = 3; }

    const float* __restrict__ W = weights + p * 1024;   // W[u*32 + w]
    const int w = wtile * 16 + lo;

    // B operands: loaded once, reused for all tiles handled by this wave.
    // B layout: VGPR v, lanes 0-15 hold K = hi*2+v (N = lane&15).
    v2f b[8];
#pragma unroll
    for (int kb = 0; kb < 8; ++kb) {
        const int k0 = kb * 4 + hi * 2;
        b[kb].x = W[(k0 + 0) * 32 + w];
        b[kb].y = W[(k0 + 1) * 32 + w];
    }

    for (int tt = 0; tt < EQ_TPW; ++tt) {
        const int tile = tgrp * EQ_TPW + tt;
        if (tile >= ntiles) break;                     // scalar-uniform

        int row = tile * 16 + lo;
        if (row >= num_nodes) row = num_nodes - 1;     // clamp loads
        const float* __restrict__ frow = features + (size_t)row * 128;

        v8f acc = {};
#pragma unroll
        for (int kb = 0; kb < 8; ++kb) {
            const int k0 = kb * 4 + hi * 2;   // A: lanes0-15 K=0,1; lanes16-31 K=2,3
            v2f a;
            a.x = frow[base + (k0 + 0) * stride];
            a.y = frow[base + (k0 + 1) * stride];
            acc = __builtin_amdgcn_wmma_f32_16x16x4_f32(
                /*neg_a=*/false, a, /*neg_b=*/false, b[kb],
                /*c_mod=*/(short)0, acc, /*reuse_a=*/false, /*reuse_b=*/false);
        }

        // D: VGPR r holds M = hi*8 + r, N = lane&15.
        // Single base pointer + immediate offsets; scalar-uniform tail check.
        float* __restrict__ zp =
            z + (size_t)(tile * 16 + hi * 8) * 352 + chunk * 32 + w;
        if (tile * 16 + 16 <= num_nodes) {             // full tile (common path)
#pragma unroll
            for (int r = 0; r < 8; ++r) zp[(size_t)r * 352] = acc[r];
        } else {                                       // ragged tail tile
#pragma unroll
            for (int r = 0; r < 8; ++r)
                if (tile * 16 + hi * 8 + r < num_nodes) zp[(size_t)r * 352] = acc[r];
        }
    }
}

// ---------------------------------------------------------------------------
// Shared per-edge epilogue: CG contraction with edge_sh + 4 fp32 atomics/lane.
// lane == output channel w.
// ---------------------------------------------------------------------------
__device__ __forceinline__ void eq_edge_store(float zA, float zB,
                                              const float zC[3], const float zD[3],
                                              const float zE[3],
                                              const float* __restrict__ sh,
                                              const Coefs& cf,
                                              float* __restrict__ ob, int lane) {
    const float s0 = sh[0];
    const float s1[3] = { sh[1], sh[2], sh[3] };
    const float s2[5] = { sh[4], sh[5], sh[6], sh[7], sh[8] };

    float t[3][3];
#pragma unroll
    for (int i = 0; i < 3; ++i)
#pragma unroll
        for (int k = 0; k < 3; ++k) {
            float acc = 0.0f;
#pragma unroll
            for (int j = 0; j < 5; ++j) acc += cf.cg4[i][j][k] * s2[j];
            t[i][k] = acc;
        }

    // scalar output block (paths 0 and 3)
    float o0 = cf.c0 * zA * s0
             + cf.c3 * (zD[0] * s1[0] + zD[1] * s1[1] + zD[2] * s1[2]);
    unsafeAtomicAdd(ob + lane, o0);

    // vector output block (paths 1, 2, 4); layout: 32 + w*3 + k
#pragma unroll
    for (int k = 0; k < 3; ++k) {
        float ov = cf.c1 * zB * s1[k]
                 + cf.c2 * s0 * zC[k]
                 + zE[0] * t[0][k] + zE[1] * t[1][k] + zE[2] * t[2][k];
        unsafeAtomicAdd(ob + 32 + lane * 3 + k, ov);
    }
}

// ---------------------------------------------------------------------------
// Kernel 2: per-edge gather of precomputed z (L2-resident) + scatter-add.
// One wave per edge; lane = w. 11 fully coalesced 128B loads per wave.
// ---------------------------------------------------------------------------
__global__ void eq_edge_from_z(const float* __restrict__ z,
                               const float* __restrict__ edge_sh,
                               const int* __restrict__ edge_index,
                               float* __restrict__ out,
                               int num_edges, Coefs cf) {
    const int wid  = blockIdx.x * (blockDim.x >> 5) + (threadIdx.x >> 5);
    const int lane = threadIdx.x & 31;
    if (wid >= num_edges) return;

    const int src = edge_index[wid];
    const int dst = edge_index[num_edges + wid];
    const float* __restrict__ zb = z + (size_t)src * 352 + lane;

    const float zA = zb[0];
    const float zB = zb[32];
    const float zC[3] = { zb[ 64], zb[ 96], zb[128] };
    const float zD[3] = { zb[160], zb[192], zb[224] };
    const float zE[3] = { zb[256], zb[288], zb[320] };

    eq_edge_store(zA, zB, zC, zD, zE,
                  edge_sh + (size_t)wid * 9, cf,
                  out + (size_t)dst * 128, lane);
}

// ---------------------------------------------------------------------------
// Fallback (workspace too small): direct per-edge weight matmul (VALU).
// ---------------------------------------------------------------------------
__global__ void eq_edge_direct(const float* __restrict__ features,
                               const float* __restrict__ edge_sh,
                               const int* __restrict__ edge_index,
                               const float* __restrict__ weights,
                               float* __restrict__ out,
                               int num_edges, Coefs cf) {
    const int wid  = blockIdx.x * (blockDim.x >> 5) + (threadIdx.x >> 5);
    const int lane = threadIdx.x & 31;
    if (wid >= num_edges) return;

    const int src = edge_index[wid];
    const int dst = edge_index[num_edges + wid];
    const float* __restrict__ frow = features + (size_t)src * 128;

    float zA = 0.f, zB = 0.f;
    float zC[3] = {0.f, 0.f, 0.f}, zD[3] = {0.f, 0.f, 0.f}, zE[3] = {0.f, 0.f, 0.f};
#pragma unroll 4
    for (int u = 0; u < 32; ++u) {
        const float x0 = frow[u];
        const float w0 = weights[0 * 1024 + u * 32 + lane];
        const float w1 = weights[1 * 1024 + u * 32 + lane];
        const float w2 = weights[2 * 1024 + u * 32 + lane];
        const float w3 = weights[3 * 1024 + u * 32 + lane];
        const float w4 = weights[4 * 1024 + u * 32 + lane];
        zA += x0 * w0;
        zB += x0 * w1;
#pragma unroll
        for (int i = 0; i < 3; ++i) {
            const float x1 = frow[32 + u * 3 + i];
            zC[i] += x1 * w2;
            zD[i] += x1 * w3;
            zE[i] += x1 * w4;
        }
    }
    eq_edge_store(zA, zB, zC, zD, zE,
                  edge_sh + (size_t)wid * 9, cf,
                  out + (size_t)dst * 128, lane);
}

// ---------------------------------------------------------------------------
extern "C" void kernel_launch(void* const* d_in, const int* in_sizes, int n_in,
                              void* d_out, int out_size, void* d_ws, size_t ws_size,
                              hipStream_t stream) {
    const float* features   = (const float*)d_in[0];
    const float* edge_sh    = (const float*)d_in[1];
    const int*   edge_index = (const int*)d_in[2];
    const float* weights    = (const float*)d_in[3];
    float*       out        = (float*)d_out;

    const int num_nodes = in_sizes[0] / 128;
    const int num_edges = in_sizes[1] / 9;

    const Coefs cf = make_coefs();

    // zero output (poisoned by harness)
    {
        const int thr = 256;
        const int blk = (out_size + thr - 1) / thr;
        eq_zero_out<<<blk, thr, 0, stream>>>(out, out_size);
    }

    const size_t zbytes = (size_t)num_nodes * 352 * sizeof(float);
    const int wavesPerBlock = 8;
    const int thr = wavesPerBlock * 32;
    const int blk = (num_edges + wavesPerBlock - 1) / wavesPerBlock;

    if (ws_size >= zbytes) {
        float* z = (float*)d_ws;
        const int ntiles = (num_nodes + 15) / 16;
        dim3 grid((ntiles + EQ_TPW - 1) / EQ_TPW, 22);
        eq_precompute_z<<<grid, 32, 0, stream>>>(features, weights, z, num_nodes);
        eq_edge_from_z<<<blk, thr, 0, stream>>>(z, edge_sh, edge_index, out,
                                                num_edges, cf);
    } else {
        eq_edge_direct<<<blk, thr, 0, stream>>>(features, edge_sh, edge_index,
                                                weights, out, num_edges, cf);
    }
}